// GraphCapsuleLayer_77884936946120
// MI455X (gfx1250) — compile-verified
//
#include <hip/hip_runtime.h>
#include <math.h>

typedef __attribute__((ext_vector_type(2))) float v2f;
typedef __attribute__((ext_vector_type(8))) float v8f;

#define NUM_CAPSULE 8
#define DIM_CAPSULE 16
#define IN_NUM_CAPS 16
#define IN_DIM_CAPS 8
#define NUM_ROUTING 3
#define CAP_EPS 1e-7f
#define SPB 16   // samples (b,n) per workgroup == WMMA M

// LDS budget: uhat [SPB][J][I][D] = 16*8*16*16 floats = 128KB
//             xs   [SPB][I][K]    = 16*16*8  floats =   8KB   -> 136KB (<320KB/WGP)
#define UHAT_FLOATS (SPB * NUM_CAPSULE * IN_NUM_CAPS * DIM_CAPSULE)
#define XS_FLOATS   (SPB * IN_NUM_CAPS * IN_DIM_CAPS)

__global__ __launch_bounds__(128) void capsule_routing_kernel(
    const float* __restrict__ x,   // [BN][I][K]
    const float* __restrict__ W,   // [J][I][D][K]
    float* __restrict__ out,       // [BN][J][D]
    int BN)
{
  extern __shared__ float smem[];
  float* uhat = smem;                 // [s][j][i][d]
  float* xs   = smem + UHAT_FLOATS;   // [s][i][k]

  const int tid = threadIdx.x;
  const int bn0 = blockIdx.x * SPB;

  // ---- stage x for 16 samples into LDS (2048 floats, coalesced float4) ----
  {
    const float4* src = (const float4*)(x + (size_t)bn0 * (IN_NUM_CAPS * IN_DIM_CAPS));
    float4* dst = (float4*)xs;
    const int n4 = XS_FLOATS / 4; // 512
    for (int t = tid; t < n4; t += 128) dst[t] = src[t];
  }
  __syncthreads();

  // ---- Phase 1: u_hat via V_WMMA_F32_16X16X4_F32 (K=8 -> 2 chained WMMAs) ----
  // Per (j,i): A = x[m=0..15, i, k]  (16x8), B = W[j,i,d,k]^T (8x16, B[k][d]),
  // C = u_hat[m, j, i, d] (16x16). Wave w handles i in [4w, 4w+4), all j.
  {
    const int wave  = tid >> 5;
    const int lane  = tid & 31;
    const int m     = lane & 15;          // A row / sample
    const int d     = lane & 15;          // B/C column
    const int hi    = lane >> 4;          // lane half
    const int kb    = hi * 2;             // A/B K sub-index per layout tables
    const int mhalf = hi * 8;             // C rows: VGPR r -> M = r + 8*hi

    for (int ii = 0; ii < 4; ++ii) {
      const int i = wave * 4 + ii;
      const float* xr = &xs[(m * IN_NUM_CAPS + i) * IN_DIM_CAPS];
      v2f a_lo, a_hi;                      // A 16x4 f32: vgpr0 K=kb, vgpr1 K=kb+1
      a_lo.x = xr[kb];     a_lo.y = xr[kb + 1];
      a_hi.x = xr[4 + kb]; a_hi.y = xr[4 + kb + 1];
      for (int j = 0; j < 8; ++j) {
        const float* wr = &W[(((j * IN_NUM_CAPS + i) * DIM_CAPSULE) + d) * IN_DIM_CAPS];
        v2f b_lo, b_hi;                    // B 4x16 f32: B[k][d] = W[j,i,d,k]
        b_lo.x = wr[kb];     b_lo.y = wr[kb + 1];
        b_hi.x = wr[4 + kb]; b_hi.y = wr[4 + kb + 1];
        v8f c = {0.f, 0.f, 0.f, 0.f, 0.f, 0.f, 0.f, 0.f};
        c = __builtin_amdgcn_wmma_f32_16x16x4_f32(false, a_lo, false, b_lo,
                                                  (short)0, c, false, false);
        c = __builtin_amdgcn_wmma_f32_16x16x4_f32(false, a_hi, false, b_hi,
                                                  (short)0, c, false, false);
#pragma unroll
        for (int r = 0; r < 8; ++r)
          uhat[(((mhalf + r) * NUM_CAPSULE + j) * IN_NUM_CAPS + i) * DIM_CAPSULE + d] = c[r];
      }
    }
  }
  __syncthreads();

  // ---- Phase 2: dynamic routing. Thread (s = tid>>3, j = tid&7). ----
  // j-threads of a sample are 8 consecutive lanes -> softmax over j via shfl_xor.
  {
    const int s = tid >> 3;
    const int j = tid & 7;
    const float* uh = &uhat[((s * NUM_CAPSULE + j) * IN_NUM_CAPS) * DIM_CAPSULE];

    float bb[IN_NUM_CAPS];
#pragma unroll
    for (int i = 0; i < IN_NUM_CAPS; ++i) bb[i] = 0.f;
    float o[DIM_CAPSULE];

    for (int it = 0; it < NUM_ROUTING; ++it) {
      // c = softmax_j(b) per (s,i)
      float cvec[IN_NUM_CAPS];
#pragma unroll
      for (int i = 0; i < IN_NUM_CAPS; ++i) {
        float v = bb[i];
        float mx = v;
        mx = fmaxf(mx, __shfl_xor(mx, 1, 32));
        mx = fmaxf(mx, __shfl_xor(mx, 2, 32));
        mx = fmaxf(mx, __shfl_xor(mx, 4, 32));
        float e = __expf(v - mx);
        float ss = e;
        ss += __shfl_xor(ss, 1, 32);
        ss += __shfl_xor(ss, 2, 32);
        ss += __shfl_xor(ss, 4, 32);
        cvec[i] = e / ss;
      }
      // outputs[d] = sum_i c[i] * uhat[s,j,i,d]
#pragma unroll
      for (int dd = 0; dd < DIM_CAPSULE; ++dd) o[dd] = 0.f;
      for (int i = 0; i < IN_NUM_CAPS; ++i) {
        const float ci = cvec[i];
        const float* row = uh + i * DIM_CAPSULE;
#pragma unroll
        for (int dd = 0; dd < DIM_CAPSULE; ++dd)
          o[dd] = fmaf(ci, row[dd], o[dd]);
      }
      if (it == NUM_ROUTING - 1) break;   // final iteration: no squash (as in source)
      // squash
      float s2 = 0.f;
#pragma unroll
      for (int dd = 0; dd < DIM_CAPSULE; ++dd) s2 = fmaf(o[dd], o[dd], s2);
      const float scale = s2 / (1.f + s2) / sqrtf(s2 + CAP_EPS);
#pragma unroll
      for (int dd = 0; dd < DIM_CAPSULE; ++dd) o[dd] *= scale;
      // b[i] += dot_d(outputs, uhat[s,j,i,:])
      for (int i = 0; i < IN_NUM_CAPS; ++i) {
        const float* row = uh + i * DIM_CAPSULE;
        float acc = 0.f;
#pragma unroll
        for (int dd = 0; dd < DIM_CAPSULE; ++dd)
          acc = fmaf(o[dd], row[dd], acc);
        bb[i] += acc;
      }
    }

    // write out[bn0+s, j, 0:16] — 16 contiguous floats per thread
    float4* op = (float4*)(out + (size_t)((bn0 + s) * NUM_CAPSULE + j) * DIM_CAPSULE);
    op[0] = make_float4(o[0],  o[1],  o[2],  o[3]);
    op[1] = make_float4(o[4],  o[5],  o[6],  o[7]);
    op[2] = make_float4(o[8],  o[9],  o[10], o[11]);
    op[3] = make_float4(o[12], o[13], o[14], o[15]);
  }
}

extern "C" void kernel_launch(void* const* d_in, const int* in_sizes, int n_in,
                              void* d_out, int out_size, void* d_ws, size_t ws_size,
                              hipStream_t stream) {
  (void)n_in; (void)d_ws; (void)ws_size; (void)out_size;
  const float* x = (const float*)d_in[0];   // [4,4000,16,8]
  const float* W = (const float*)d_in[1];   // [8,16,16,8]
  float* out = (float*)d_out;               // [4,4000,8,16]

  const int BN = in_sizes[0] / (IN_NUM_CAPS * IN_DIM_CAPS);  // 16000
  const int nblocks = BN / SPB;                              // 1000
  const size_t shbytes = (size_t)(UHAT_FLOATS + XS_FLOATS) * sizeof(float); // 136KB

  static int attr_set = 0;  // attribute is idempotent config, not state affecting output
  if (!attr_set) {
    hipFuncSetAttribute((const void*)capsule_routing_kernel,
                        hipFuncAttributeMaxDynamicSharedMemorySize, (int)shbytes);
    attr_set = 1;
  }

  capsule_routing_kernel<<<nblocks, 128, shbytes, stream>>>(x, W, out, BN);
}